// SplineNet_56831007261230
// MI455X (gfx1250) — compile-verified
//
#include <hip/hip_runtime.h>
#include <hip/hip_bf16.h>
#include <math.h>

typedef __attribute__((ext_vector_type(2))) float v2f;
typedef __attribute__((ext_vector_type(8))) float v8f;

#define DFEAT 128
#define TILE_K 32
#define LDS_STRIDE 144   // 16-float pad => half-wave B reads hit disjoint bank groups

// ---------------------------------------------------------------- zero fill
__global__ void zero_f32x4(float4* __restrict__ p, long long n4) {
  long long i = (long long)blockIdx.x * blockDim.x + threadIdx.x;
  long long stride = (long long)gridDim.x * blockDim.x;
  float4 z = make_float4(0.f, 0.f, 0.f, 0.f);
  for (; i < n4; i += stride) p[i] = z;
}

// ---------------------------------------------------------------- degree
__global__ void degree_kernel(const int* __restrict__ dst, float* __restrict__ deg, int E) {
  int i = blockIdx.x * blockDim.x + threadIdx.x;
  int stride = gridDim.x * blockDim.x;
  for (; i < E; i += stride) atomicAdd(&deg[dst[i]], 1.0f);
}

__global__ void rdeg_kernel(const float* __restrict__ deg, float* __restrict__ rdg, int N) {
  int i = blockIdx.x * blockDim.x + threadIdx.x;
  if (i < N) rdg[i] = 1.0f / fmaxf(deg[i], 1.0f);
}

// ------------------------------------------------- edge scatter (wave/edge)
__global__ void scatter_kernel(const float* __restrict__ x,
                               const int* __restrict__ src,
                               const int* __restrict__ dst,
                               const float* __restrict__ attr,
                               float* __restrict__ acc, int K, int E) {
  int e = (int)(((long long)blockIdx.x * blockDim.x + threadIdx.x) >> 5);
  if (e >= E) return;
  int lane = threadIdx.x & 31;
  int s = src[e], d = dst[e];
  float v = attr[e] * (float)(K - 1);
  float i0f = floorf(v);
  float fr = v - i0f;
  int i0 = (int)i0f;
  i0 = i0 < 0 ? 0 : (i0 > K - 1 ? K - 1 : i0);
  int i1 = (i0 + 1 > K - 1) ? K - 1 : i0 + 1;
  const float4 xs = *(const float4*)(x + (size_t)s * DFEAT + lane * 4);
  float w0 = 1.0f - fr, w1 = fr;
  float* p0 = acc + ((size_t)d * K + i0) * DFEAT + lane * 4;
  float* p1 = acc + ((size_t)d * K + i1) * DFEAT + lane * 4;
  atomicAdd(p0 + 0, w0 * xs.x); atomicAdd(p0 + 1, w0 * xs.y);
  atomicAdd(p0 + 2, w0 * xs.z); atomicAdd(p0 + 3, w0 * xs.w);
  atomicAdd(p1 + 0, w1 * xs.x); atomicAdd(p1 + 1, w1 * xs.y);
  atomicAdd(p1 + 2, w1 * xs.z); atomicAdd(p1 + 3, w1 * xs.w);
}

// ------------------------------------------ fused WMMA GEMM + bias + ELU
// out[N,128] = ELU( (acc/deg) @ W  +  x @ root  + bias )
// A = [acc_row * rdeg | x_row]   (Ktot = KD + 128)
// B = [W ; root]   double-buffered in LDS via GLOBAL_LOAD_ASYNC_TO_LDS_B128
__global__ __launch_bounds__(256) void gemm_elu_kernel(
    const float* __restrict__ acc, const float* __restrict__ xin,
    const float* __restrict__ W, const float* __restrict__ root,
    const float* __restrict__ bias, const float* __restrict__ rdg,
    float* __restrict__ out, int KD, int N) {
  __shared__ float Bt[2][TILE_K * LDS_STRIDE];
  const int Ktot = KD + DFEAT;
  const int nchunks = Ktot / TILE_K;
  const int tid  = threadIdx.x;
  const int wave = tid >> 5;
  const int lane = tid & 31;
  const int hf   = lane >> 4;        // half-wave select (K offset +2 / M offset +8)
  const int lrow = lane & 15;
  const int rowbase = blockIdx.x * 128 + wave * 16;
  int arow = rowbase + lrow;
  if (arow > N - 1) arow = N - 1;    // clamp loads; stores guarded below
  const float rd = rdg[arow];

  v8f c[8] = {};                     // 8 N-tiles of 16x16 f32 accumulators

  // async stage of B[k0:k0+32, 0:128] into LDS buffer `buf` (no VGPR transit)
  auto stage_async = [&](int chunk, int buf) {
#pragma unroll
    for (int ii = 0; ii < 4; ++ii) {
      int i = tid + ii * 256;                      // 1024 float4 tiles total
      int r = i >> 5;
      int cc = (i & 31) * 4;
      int gr = chunk * TILE_K + r;
      const float* sp = (gr < KD) ? (W + (size_t)gr * DFEAT + cc)
                                  : (root + (size_t)(gr - KD) * DFEAT + cc);
      unsigned lds = (unsigned)(uintptr_t)(&Bt[buf][r * LDS_STRIDE + cc]);
      unsigned long long ga = (unsigned long long)(uintptr_t)sp;
      asm volatile("global_load_async_to_lds_b128 %0, %1, off"
                   :: "v"(lds), "v"(ga) : "memory");
    }
  };

  stage_async(0, 0);
  asm volatile("s_wait_asynccnt 0x0" ::: "memory");
  __syncthreads();

  int cur = 0;
  for (int chunk = 0; chunk < nchunks; ++chunk) {
    if (chunk + 1 < nchunks) stage_async(chunk + 1, cur ^ 1);  // prefetch next B tile

    const int k0 = chunk * TILE_K;
    const bool inAcc = (k0 < KD);    // KD is a multiple of TILE_K => chunk-uniform
    const float* Ap = inAcc ? (acc + (size_t)arow * KD + k0)
                            : (xin + (size_t)arow * DFEAT + (k0 - KD));
    const float scale = inAcc ? rd : 1.0f;
    const float* B0 = &Bt[cur][0];

    // prefetch the whole chunk's A fragments: 8 independent b64 loads in flight
    v2f a[8];
#pragma unroll
    for (int j = 0; j < 8; ++j)
      a[j] = *(const v2f*)(Ap + j * 4 + hf * 2);

#pragma unroll
    for (int j = 0; j < 8; ++j) {
      const int kk = j * 4;
      // A fragment: lanes 0-15 -> K = kk,kk+1 ; lanes 16-31 -> K = kk+2,kk+3
      v2f av = a[j];
      av.x *= scale;
      av.y *= scale;
#pragma unroll
      for (int n = 0; n < 8; ++n) {
        v2f b;
        b.x = B0[(kk + hf * 2)     * LDS_STRIDE + n * 16 + lrow];
        b.y = B0[(kk + hf * 2 + 1) * LDS_STRIDE + n * 16 + lrow];
        c[n] = __builtin_amdgcn_wmma_f32_16x16x4_f32(
            false, av, false, b, (short)0, c[n], false, false);
      }
    }

    asm volatile("s_wait_asynccnt 0x0" ::: "memory");  // own async ops done
    __syncthreads();                                   // collectively done
    cur ^= 1;
  }

  // epilogue: bias + ELU, D-tile layout: VGPR i, lanes0-15 -> M=i, lanes16-31 -> M=8+i
#pragma unroll
  for (int n = 0; n < 8; ++n) {
    int col = n * 16 + lrow;
    float bv = bias[col];
#pragma unroll
    for (int i = 0; i < 8; ++i) {
      int r = rowbase + i + hf * 8;
      if (r < N) {
        float val = c[n][i] + bv;
        val = val > 0.0f ? val : (__expf(val) - 1.0f);  // ELU(alpha=1)
        out[(size_t)r * DFEAT + col] = val;
      }
    }
  }
}

// ---------------------------------------------------------------- head
__global__ void head_kernel(const float* __restrict__ h, const float* __restrict__ Wm,
                            const float* __restrict__ bm, float* __restrict__ out, int N) {
  int row = (int)(((long long)blockIdx.x * blockDim.x + threadIdx.x) >> 5);
  int lane = threadIdx.x & 31;
  if (row >= N) return;
  float c0 = 0.f, c1 = 0.f;
  for (int j = lane; j < DFEAT; j += 32) {
    float hv = h[(size_t)row * DFEAT + j];
    c0 += hv * Wm[j * 2 + 0];
    c1 += hv * Wm[j * 2 + 1];
  }
  for (int off = 16; off > 0; off >>= 1) {
    c0 += __shfl_down(c0, off);
    c1 += __shfl_down(c1, off);
  }
  if (lane == 0) {
    float o0 = c0 + bm[0], o1 = c1 + bm[1];
    out[row * 2 + 0] = o0 > 0.f ? o0 : 0.f;
    out[row * 2 + 1] = o1 > 0.f ? o1 : 0.f;
  }
}

// ---------------------------------------------------------------- launch
extern "C" void kernel_launch(void* const* d_in, const int* in_sizes, int n_in,
                              void* d_out, int out_size, void* d_ws, size_t ws_size,
                              hipStream_t stream) {
  const float* x     = (const float*)d_in[0];
  const int*   eidx  = (const int*)  d_in[1];
  const float* attr  = (const float*)d_in[2];
  const float* W1    = (const float*)d_in[3];
  const float* root1 = (const float*)d_in[4];
  const float* b1    = (const float*)d_in[5];
  const float* W2    = (const float*)d_in[6];
  const float* root2 = (const float*)d_in[7];
  const float* b2    = (const float*)d_in[8];
  const float* Wm    = (const float*)d_in[9];
  const float* bm    = (const float*)d_in[10];
  float* out = (float*)d_out;

  const int N  = in_sizes[0] / DFEAT;
  const int E  = in_sizes[1] / 2;
  const int K1 = in_sizes[3] / (DFEAT * DFEAT);
  const int K2 = in_sizes[6] / (DFEAT * DFEAT);
  const int* src = eidx;
  const int* dst = eidx + E;

  // workspace layout (256B aligned): deg, rdeg, acc(max K2), h1, h2
  char* ws = (char*)d_ws;
  size_t off = 0;
  auto take = [&](size_t bytes) {
    void* p = ws + off;
    off = (off + bytes + 255) & ~(size_t)255;
    return p;
  };
  float* deg  = (float*)take((size_t)N * 4);
  float* rdg  = (float*)take((size_t)N * 4);
  float* accb = (float*)take((size_t)N * K2 * DFEAT * 4);
  float* h1   = (float*)take((size_t)N * DFEAT * 4);
  float* h2   = (float*)take((size_t)N * DFEAT * 4);
  (void)ws_size; (void)n_in; (void)out_size;

  const int ZB = 256;
  auto zgrid = [](long long n4) { long long g = (n4 + 255) / 256; return (int)(g > 16384 ? 16384 : (g < 1 ? 1 : g)); };

  // --- degree (once; shared by both layers) ---
  long long degN4 = (long long)N / 4;
  zero_f32x4<<<zgrid(degN4), ZB, 0, stream>>>((float4*)deg, degN4);
  degree_kernel<<<dim3((E + 255) / 256 > 8192 ? 8192 : (E + 255) / 256), 256, 0, stream>>>(dst, deg, E);
  rdeg_kernel<<<dim3((N + 255) / 256), 256, 0, stream>>>(deg, rdg, N);

  // --- layer 1 ---
  long long acc1N4 = (long long)N * K1 * DFEAT / 4;
  zero_f32x4<<<zgrid(acc1N4), ZB, 0, stream>>>((float4*)accb, acc1N4);
  scatter_kernel<<<dim3((E + 7) / 8), 256, 0, stream>>>(x, src, dst, attr, accb, K1, E);
  gemm_elu_kernel<<<dim3((N + 127) / 128), 256, 0, stream>>>(
      accb, x, W1, root1, b1, rdg, h1, K1 * DFEAT, N);

  // --- layer 2 ---
  long long acc2N4 = (long long)N * K2 * DFEAT / 4;
  zero_f32x4<<<zgrid(acc2N4), ZB, 0, stream>>>((float4*)accb, acc2N4);
  scatter_kernel<<<dim3((E + 7) / 8), 256, 0, stream>>>(h1, src, dst, attr, accb, K2, E);
  gemm_elu_kernel<<<dim3((N + 127) / 128), 256, 0, stream>>>(
      accb, h1, W2, root2, b2, rdg, h2, K2 * DFEAT, N);

  // --- head ---
  head_kernel<<<dim3(((long long)N * 32 + 255) / 256), 256, 0, stream>>>(h2, Wm, bm, out, N);
}